// DeltaMemoryLayer_73100343378557
// MI455X (gfx1250) — compile-verified
//
#include <hip/hip_runtime.h>
#include <hip/hip_bf16.h>

#define HID   1024
#define MEMD  64
#define BATCH 4
#define SEQ   4096
#define BL    (BATCH*SEQ)

typedef float v2f __attribute__((ext_vector_type(2)));
typedef float v8f __attribute__((ext_vector_type(8)));
typedef unsigned int u32x4 __attribute__((ext_vector_type(4)));
typedef int i32x4 __attribute__((ext_vector_type(4)));
typedef int i32x8 __attribute__((ext_vector_type(8)));

#define ASTR 68    // A-tile LDS stride (TDM pad: 4 DWORDs after every 64 DWORDs)
#define PSTR 160   // B-tile pair-row stride in floats (160%64==32 -> disjoint banks for half-waves)

// ---------------------------------------------------------------------------
// TDM: DMA a 64x64 f32 tile (row pitch = row_stride_elems) into LDS with
// 4-DWORD padding every 64 DWORDs (=> LDS row stride ASTR=68 floats).
// D# packing per CDNA5 ISA 8.3/8.4:
//   g0: [1:0]=count=1, [63:32]=lds_addr, [120:64]=global_addr, [127:126]=type=2
//   g1: [17:16]=data_size(2=4B), [20]=pad_en, [24:22]=pad_interval(5 => 64 DW),
//       [31:25]=pad_amount(3 => 4 DW), [79:48]=tensor_dim0=64, [111:80]=tensor_dim1=64,
//       [127:112]=tile_dim0=64, [143:128]=tile_dim1=64, [207:160]=tensor_dim0_stride
// ---------------------------------------------------------------------------
__device__ __forceinline__ void tdm_load_tile64x64(unsigned lds_byte_off,
                                                   const void* gptr,
                                                   unsigned row_stride_elems)
{
    unsigned long long ga = (unsigned long long)(uintptr_t)gptr;
    u32x4 g0;
    g0[0] = 1u;                                                  // count=1, user mode
    g0[1] = lds_byte_off;                                        // lds_addr
    g0[2] = (unsigned)ga;                                        // global_addr[31:0]
    g0[3] = (unsigned)((ga >> 32) & 0x01FFFFFFu) | (2u << 30);   // addr[56:32] | type=2
    i32x8 g1;
    g1[0] = (2 << 16) | (1 << 20) | (5 << 22) | (3 << 25);       // 4B, pad_en, 64DW interval, +4DW
    g1[1] = 64 << 16;                                            // tensor_dim0[15:0] @ bits 63:48
    g1[2] = 64 << 16;                                            // tensor_dim1[15:0] @ bits 95:80
    g1[3] = 64 << 16;                                            // tile_dim0 @ bits 127:112
    g1[4] = 64;                                                  // tile_dim1 @ bits 143:128
    g1[5] = (int)row_stride_elems;                               // tensor_dim0_stride[31:0]
    g1[6] = 0;
    g1[7] = 0;
    i32x4 z = {0, 0, 0, 0};
#if __clang_major__ >= 23
    i32x8 z8 = {0, 0, 0, 0, 0, 0, 0, 0};
    __builtin_amdgcn_tensor_load_to_lds(g0, g1, z, z, z8, 0);
#else
    __builtin_amdgcn_tensor_load_to_lds(g0, g1, z, z, 0);
#endif
}

// ---------------------------------------------------------------------------
// Kernel 1: fused QKV projection   Q|K|V = x @ Wq|Wk|Wv   (fp32 WMMA 16x16x4)
// block = 128 (4 waves); each wave computes a 16(M) x 64(N) strip.
// A tile arrives via TDM (wave 0) while all threads fill the pair-interleaved
// B tile; B operand per WMMA is then a single aligned ds_load_b64.
// ---------------------------------------------------------------------------
__global__ __launch_bounds__(128) void qkv_kernel(
    const float* __restrict__ x,
    const float* __restrict__ Wq, const float* __restrict__ Wk,
    const float* __restrict__ Wv,
    float* __restrict__ Qo, float* __restrict__ Ko, float* __restrict__ Vo)
{
    __shared__ __align__(16) float tA[64 * ASTR];   // 64 rows x 64 K (stride 68 via TDM pad)
    __shared__ __align__(16) float tB[32 * PSTR];   // 32 K-pairs x 64 N x 2 (pair-interleaved)

    const int tid  = threadIdx.x;
    const int wid  = tid >> 5;
    const int lane = tid & 31;
    const int half = lane >> 4;     // 0: K+{0,1}, 1: K+{2,3}
    const int lr   = lane & 15;
    const int mb   = blockIdx.x * 64;

    const float* W;
    float* Out;
    if (blockIdx.y == 0)      { W = Wq; Out = Qo; }
    else if (blockIdx.y == 1) { W = Wk; Out = Ko; }
    else                      { W = Wv; Out = Vo; }

    const unsigned ldsA = (unsigned)(uintptr_t)(void*)tA;

    v8f acc0 = {}, acc1 = {}, acc2 = {}, acc3 = {};

    for (int kb = 0; kb < HID; kb += 64) {
        __syncthreads();  // all waves done reading previous tiles
        if (wid == 0)     // one TDM issue per workgroup (wave-uniform branch)
            tdm_load_tile64x64(ldsA, x + (size_t)mb * HID + kb, HID);
        // pair-interleaved B fill (overlaps the TDM)
        for (int j = 0; j < 32; ++j) {
            int e = tid + 128 * j;
            int r = e >> 6, c = e & 63;                    // r = local k, c = n
            tB[(r >> 1) * PSTR + c * 2 + (r & 1)] = W[(kb + r) * MEMD + c];
        }
        if (wid == 0)
            __builtin_amdgcn_s_wait_tensorcnt(0);
        __syncthreads();

        const int row = wid * 16 + lr;
        #pragma unroll
        for (int kk = 0; kk < 64; kk += 4) {
            const int kA = kk + 2 * half;                  // A: K offset for this half-wave
            const int p  = (kk >> 1) + half;               // B: pair row index
            v2f a = *(const v2f*)&tA[row * ASTR + kA];
            v2f b;
            b = *(const v2f*)&tB[p * PSTR + (0 * 16 + lr) * 2];
            acc0 = __builtin_amdgcn_wmma_f32_16x16x4_f32(false, a, false, b, (short)0, acc0, false, false);
            b = *(const v2f*)&tB[p * PSTR + (1 * 16 + lr) * 2];
            acc1 = __builtin_amdgcn_wmma_f32_16x16x4_f32(false, a, false, b, (short)0, acc1, false, false);
            b = *(const v2f*)&tB[p * PSTR + (2 * 16 + lr) * 2];
            acc2 = __builtin_amdgcn_wmma_f32_16x16x4_f32(false, a, false, b, (short)0, acc2, false, false);
            b = *(const v2f*)&tB[p * PSTR + (3 * 16 + lr) * 2];
            acc3 = __builtin_amdgcn_wmma_f32_16x16x4_f32(false, a, false, b, (short)0, acc3, false, false);
        }
    }

    // C/D layout: VGPR j -> M=j (lanes 0-15) / M=8+j (lanes 16-31), N=lr
    const int mrow = mb + wid * 16 + half * 8;
    #pragma unroll
    for (int j = 0; j < 8; ++j) {
        Out[(mrow + j) * MEMD + 0 * 16 + lr] = acc0[j];
        Out[(mrow + j) * MEMD + 1 * 16 + lr] = acc1[j];
        Out[(mrow + j) * MEMD + 2 * 16 + lr] = acc2[j];
        Out[(mrow + j) * MEMD + 3 * 16 + lr] = acc3[j];
    }
}

// ---------------------------------------------------------------------------
// wave32 XOR-butterfly sum via ds_swizzle (group-of-32: offset = xor<<10 | 0x1F)
// ---------------------------------------------------------------------------
__device__ __forceinline__ float wave_reduce_add(float v)
{
    v += __int_as_float(__builtin_amdgcn_ds_swizzle(__float_as_int(v), 0x041F)); // xor 1
    v += __int_as_float(__builtin_amdgcn_ds_swizzle(__float_as_int(v), 0x081F)); // xor 2
    v += __int_as_float(__builtin_amdgcn_ds_swizzle(__float_as_int(v), 0x101F)); // xor 4
    v += __int_as_float(__builtin_amdgcn_ds_swizzle(__float_as_int(v), 0x201F)); // xor 8
    v += __int_as_float(__builtin_amdgcn_ds_swizzle(__float_as_int(v), 0x401F)); // xor 16
    return v;
}

// ---------------------------------------------------------------------------
// Kernel 2: sequential delta-rule scan. One block per batch, 64 threads (2 waves).
// Thread d owns mem row d in 64 VGPRs; q/k/v staged in LDS in chunks of 16 steps;
// next chunk prefetched (per-lane addresses 64B apart cover the 4KB span).
// ---------------------------------------------------------------------------
#define CH 16
__global__ __launch_bounds__(64) void scan_kernel(
    const float* __restrict__ Q, const float* __restrict__ K,
    const float* __restrict__ V, const float* __restrict__ beta_param,
    float* __restrict__ S, float* __restrict__ final_mem)
{
    __shared__ __align__(16) float sQ[CH * 64];
    __shared__ __align__(16) float sK[CH * 64];
    __shared__ __align__(16) float sV[CH * 64];
    __shared__ float swred[2];

    const int b    = blockIdx.x;
    const int d    = threadIdx.x;
    const int wid  = d >> 5;
    const int lane = d & 31;

    const float bp   = beta_param[0];
    const float beta = 0.1f + 0.8f * (1.0f / (1.0f + __expf(-bp)));

    float m[64];
    #pragma unroll
    for (int e = 0; e < 64; ++e) m[e] = 0.0f;

    const size_t base = (size_t)b * SEQ * MEMD;

    for (int t0 = 0; t0 < SEQ; t0 += CH) {
        __syncthreads();
        const float4* gq = (const float4*)(Q + base + (size_t)t0 * MEMD);
        const float4* gk = (const float4*)(K + base + (size_t)t0 * MEMD);
        const float4* gv = (const float4*)(V + base + (size_t)t0 * MEMD);
        float4* lq = (float4*)sQ;
        float4* lk = (float4*)sK;
        float4* lv = (float4*)sV;
        #pragma unroll
        for (int j = 0; j < 4; ++j) {
            int idx = j * 64 + d;   // CH*64/4 = 256 float4 per array
            lq[idx] = gq[idx];
            lk[idx] = gk[idx];
            lv[idx] = gv[idx];
        }
        if (t0 + CH < SEQ) {        // prefetch next chunk: 64 lanes x 64B lines
            const size_t nb_ = base + (size_t)(t0 + CH) * MEMD + d * 16;
            __builtin_prefetch(Q + nb_, 0, 0);
            __builtin_prefetch(K + nb_, 0, 0);
            __builtin_prefetch(V + nb_, 0, 0);
        }
        __syncthreads();

        for (int tt = 0; tt < CH; ++tt) {
            const float* kp = &sK[tt * 64];
            const float* qp = &sQ[tt * 64];

            float vp = 0.0f, kk2 = 0.0f;
            #pragma unroll 8
            for (int e = 0; e < 64; ++e) {
                float kv = kp[e];
                vp  = __fmaf_rn(m[e], kv, vp);
                kk2 = __fmaf_rn(kv, kv, kk2);
            }
            float delta = sV[tt * 64 + d] - vp;

            float dd2 = wave_reduce_add(delta * delta);
            if (lane == 0) swred[wid] = dd2;
            __syncthreads();
            dd2 = swred[0] + swred[1];
            __syncthreads();

            // ||delta (x) k||_F = ||delta|| * ||k||
            float nrm = sqrtf(dd2 * kk2);
            float sf  = fminf(1.0f, 50.0f / (nrm + 1e-8f));
            float c   = beta * sf * delta;

            float o = 0.0f;
            #pragma unroll 8
            for (int e = 0; e < 64; ++e) {
                float kv = kp[e];
                float me = __fmaf_rn(c, kv, m[e]);
                m[e] = me;
                o = __fmaf_rn(me, qp[e], o);
            }
            S[base + (size_t)(t0 + tt) * MEMD + d] = o;
        }
    }

    #pragma unroll 4
    for (int e = 0; e < 64; ++e)
        final_mem[((size_t)b * MEMD + d) * MEMD + e] = m[e];
}

// ---------------------------------------------------------------------------
// Kernel 3: output projection  out = (S @ Wo + bo) * |memory_scale|
// grid = (BL/64, HID/64), block = 128, K = 64 (single tile; A via TDM).
// ---------------------------------------------------------------------------
__global__ __launch_bounds__(128) void out_kernel(
    const float* __restrict__ S, const float* __restrict__ Wo,
    const float* __restrict__ bo, const float* __restrict__ ms,
    float* __restrict__ Out)
{
    __shared__ __align__(16) float tA[64 * ASTR];   // S tile: 64 rows x K=64
    __shared__ __align__(16) float tB[32 * PSTR];   // Wo tile: pair-interleaved

    const int tid  = threadIdx.x;
    const int wid  = tid >> 5;
    const int lane = tid & 31;
    const int half = lane >> 4;
    const int lr   = lane & 15;
    const int mb   = blockIdx.x * 64;
    const int nb   = blockIdx.y * 64;

    if (wid == 0)
        tdm_load_tile64x64((unsigned)(uintptr_t)(void*)tA,
                           S + (size_t)mb * MEMD, MEMD);
    for (int j = 0; j < 32; ++j) {
        int e = tid + 128 * j;
        int r = e >> 6, c = e & 63;                 // r = k, c = local n
        tB[(r >> 1) * PSTR + c * 2 + (r & 1)] = Wo[r * HID + nb + c];
    }
    if (wid == 0)
        __builtin_amdgcn_s_wait_tensorcnt(0);
    __syncthreads();

    v8f acc0 = {}, acc1 = {}, acc2 = {}, acc3 = {};
    const int row = wid * 16 + lr;
    #pragma unroll
    for (int kk = 0; kk < 64; kk += 4) {
        const int kA = kk + 2 * half;
        const int p  = (kk >> 1) + half;
        v2f a = *(const v2f*)&tA[row * ASTR + kA];
        v2f b;
        b = *(const v2f*)&tB[p * PSTR + (0 * 16 + lr) * 2];
        acc0 = __builtin_amdgcn_wmma_f32_16x16x4_f32(false, a, false, b, (short)0, acc0, false, false);
        b = *(const v2f*)&tB[p * PSTR + (1 * 16 + lr) * 2];
        acc1 = __builtin_amdgcn_wmma_f32_16x16x4_f32(false, a, false, b, (short)0, acc1, false, false);
        b = *(const v2f*)&tB[p * PSTR + (2 * 16 + lr) * 2];
        acc2 = __builtin_amdgcn_wmma_f32_16x16x4_f32(false, a, false, b, (short)0, acc2, false, false);
        b = *(const v2f*)&tB[p * PSTR + (3 * 16 + lr) * 2];
        acc3 = __builtin_amdgcn_wmma_f32_16x16x4_f32(false, a, false, b, (short)0, acc3, false, false);
    }

    const float scale = fabsf(ms[0]);
    const int mrow = mb + wid * 16 + half * 8;
    #pragma unroll
    for (int j = 0; j < 8; ++j) {
        const int rowbase = (mrow + j) * HID + nb;
        Out[rowbase + 0 * 16 + lr] = (acc0[j] + bo[nb + 0 * 16 + lr]) * scale;
        Out[rowbase + 1 * 16 + lr] = (acc1[j] + bo[nb + 1 * 16 + lr]) * scale;
        Out[rowbase + 2 * 16 + lr] = (acc2[j] + bo[nb + 2 * 16 + lr]) * scale;
        Out[rowbase + 3 * 16 + lr] = (acc3[j] + bo[nb + 3 * 16 + lr]) * scale;
    }
}

// ---------------------------------------------------------------------------
extern "C" void kernel_launch(void* const* d_in, const int* in_sizes, int n_in,
                              void* d_out, int out_size, void* d_ws, size_t ws_size,
                              hipStream_t stream)
{
    const float* x  = (const float*)d_in[0];
    const float* Wq = (const float*)d_in[1];
    const float* Wk = (const float*)d_in[2];
    const float* Wv = (const float*)d_in[3];
    const float* Wo = (const float*)d_in[4];
    const float* bo = (const float*)d_in[5];
    const float* ms = (const float*)d_in[6];
    const float* bp = (const float*)d_in[7];

    float* out = (float*)d_out;
    float* Qw  = (float*)d_ws;
    float* Kw  = Qw + (size_t)BL * MEMD;
    float* Vw  = Kw + (size_t)BL * MEMD;
    float* Sw  = Vw + (size_t)BL * MEMD;
    float* final_mem = out + (size_t)BL * HID;  // outputs first, then final_mem

    qkv_kernel<<<dim3(BL / 64, 3), 128, 0, stream>>>(x, Wq, Wk, Wv, Qw, Kw, Vw);
    scan_kernel<<<dim3(BATCH), 64, 0, stream>>>(Qw, Kw, Vw, bp, Sw, final_mem);
    out_kernel<<<dim3(BL / 64, HID / 64), 128, 0, stream>>>(Sw, Wo, bo, ms, out);
}